// SemanticFrameProcessingUnit_11235634446445
// MI455X (gfx1250) — compile-verified
//
#include <hip/hip_runtime.h>
#include <stdint.h>
#include <stddef.h>

// Problem constants (match reference)
#define Nn 50000
#define Ee 500000
#define Dd 128
#define Hh 8
#define DHh 16
#define Kk 400000   // 0.8 * E

typedef __attribute__((ext_vector_type(2))) float v2f;
typedef __attribute__((ext_vector_type(8))) float v8f;

// ---------------- ordered-int float encoding (for atomic max / histogram) ---
__device__ __forceinline__ unsigned f2ord(float f) {
  unsigned u = __float_as_uint(f);
  return u ^ (unsigned)(((int)u >> 31) | 0x80000000);
}
__device__ __forceinline__ float ord2f(unsigned o) {
  unsigned u = (o & 0x80000000u) ? (o & 0x7fffffffu) : ~o;
  return __uint_as_float(u);
}

// ---------------- BatchNorm statistics ------------------------------------
// 128 threads/block, each thread owns one column; rows strided over blocks.
__global__ void k_bn_stats(const float* __restrict__ x, float* __restrict__ sums) {
  int d = threadIdx.x;
  float s = 0.f, s2 = 0.f;
  for (int r = blockIdx.x; r < Nn; r += gridDim.x) {
    float v = x[r * Dd + d];
    s += v; s2 += v * v;
  }
  atomicAdd(&sums[d], s);
  atomicAdd(&sums[Dd + d], s2);
}

// Fold BN into affine: xn = x*scale + shift
__global__ void k_bn_fin(const float* __restrict__ sums, const float* __restrict__ g,
                         const float* __restrict__ beta, float* __restrict__ scsh) {
  int d = threadIdx.x;
  float mu  = sums[d] / (float)Nn;
  float var = sums[Dd + d] / (float)Nn - mu * mu;
  float rs  = rsqrtf(var + 1e-5f);
  float sc  = rs * g[d];
  scsh[d]      = sc;
  scsh[Dd + d] = beta[d] - mu * sc;
}

// ---------------- fp32 WMMA GEMMs -----------------------------------------
// Block = 256 threads = 8 wave32. Each wave owns a 16x16 output tile (one head
// worth of columns); block covers 16 rows x 128 cols. A tile staged in LDS.

// h = (x*scale+shift) @ W    [rows x 128] @ [128 x 128]
__global__ void k_node_proj(const float* __restrict__ x, const float* __restrict__ W,
                            const float* __restrict__ scsh, float* __restrict__ h) {
  __shared__ float As[16 * Dd];
  int row0 = blockIdx.x * 16;
  for (int i = threadIdx.x; i < 16 * Dd; i += 256) {
    int rr = i >> 7, kk = i & 127;
    As[i] = x[(row0 + rr) * Dd + kk] * scsh[kk] + scsh[Dd + kk];
  }
  __syncthreads();
  int lane = threadIdx.x & 31, wv = threadIdx.x >> 5;
  int col0 = wv * 16, r = lane & 15, kh = (lane >> 4) * 2;
  v8f c = {};
  for (int k0 = 0; k0 < Dd; k0 += 4) {
    int ka = k0 + kh;
    v2f a, b;
    a.x = As[r * Dd + ka];
    a.y = As[r * Dd + ka + 1];
    b.x = W[ka * Dd + col0 + r];
    b.y = W[(ka + 1) * Dd + col0 + r];
    c = __builtin_amdgcn_wmma_f32_16x16x4_f32(false, a, false, b, (short)0, c, false, false);
  }
  int rb = (lane >> 4) << 3;
  for (int v = 0; v < 8; ++v)
    h[(row0 + rb + v) * Dd + col0 + r] = c[v];
}

// he = ea @ We  (no BN)
__global__ void k_edge_proj(const float* __restrict__ ea, const float* __restrict__ We,
                            float* __restrict__ he) {
  __shared__ float As[16 * Dd];
  int row0 = blockIdx.x * 16;
  for (int i = threadIdx.x; i < 16 * Dd; i += 256) {
    int rr = i >> 7, kk = i & 127;
    As[i] = ea[(size_t)(row0 + rr) * Dd + kk];
  }
  __syncthreads();
  int lane = threadIdx.x & 31, wv = threadIdx.x >> 5;
  int col0 = wv * 16, r = lane & 15, kh = (lane >> 4) * 2;
  v8f c = {};
  for (int k0 = 0; k0 < Dd; k0 += 4) {
    int ka = k0 + kh;
    v2f a, b;
    a.x = As[r * Dd + ka];
    a.y = As[r * Dd + ka + 1];
    b.x = We[ka * Dd + col0 + r];
    b.y = We[(ka + 1) * Dd + col0 + r];
    c = __builtin_amdgcn_wmma_f32_16x16x4_f32(false, a, false, b, (short)0, c, false, false);
  }
  int rb = (lane >> 4) << 3;
  for (int v = 0; v < 8; ++v)
    he[(size_t)(row0 + rb + v) * Dd + col0 + r] = c[v];
}

// f = [xa|xb] @ Wf + bf   (K = 256)
__global__ void k_fusion(const float* __restrict__ xa, const float* __restrict__ xb,
                         const float* __restrict__ Wf, const float* __restrict__ bf,
                         float* __restrict__ f) {
  __shared__ float As[16 * 256];
  int row0 = blockIdx.x * 16;
  for (int i = threadIdx.x; i < 16 * 256; i += 256) {
    int rr = i >> 8, kk = i & 255;
    As[i] = (kk < Dd) ? xa[(row0 + rr) * Dd + kk] : xb[(row0 + rr) * Dd + kk - Dd];
  }
  __syncthreads();
  int lane = threadIdx.x & 31, wv = threadIdx.x >> 5;
  int col0 = wv * 16, r = lane & 15, kh = (lane >> 4) * 2;
  v8f c = {};
  for (int k0 = 0; k0 < 256; k0 += 4) {
    int ka = k0 + kh;
    v2f a, b;
    a.x = As[r * 256 + ka];
    a.y = As[r * 256 + ka + 1];
    b.x = Wf[ka * Dd + col0 + r];
    b.y = Wf[(ka + 1) * Dd + col0 + r];
    c = __builtin_amdgcn_wmma_f32_16x16x4_f32(false, a, false, b, (short)0, c, false, false);
  }
  int rb = (lane >> 4) << 3;
  float bv = bf[col0 + r];
  for (int v = 0; v < 8; ++v)
    f[(row0 + rb + v) * Dd + col0 + r] = c[v] + bv;
}

// ---------------- attention ------------------------------------------------
// per (node, head): dot of h slice with a_src / a_dst
__global__ void k_node_scores(const float* __restrict__ h, const float* __restrict__ asrc,
                              const float* __restrict__ adst, float* __restrict__ ssrc,
                              float* __restrict__ sdst) {
  int t = blockIdx.x * blockDim.x + threadIdx.x;
  if (t >= Nn * Hh) return;
  int node = t >> 3, hd = t & 7;
  const float* hp = h + node * Dd + hd * DHh;
  float s1 = 0.f, s2 = 0.f;
#pragma unroll
  for (int d = 0; d < DHh; ++d) {
    float v = hp[d];
    s1 += v * asrc[hd * DHh + d];
    s2 += v * adst[hd * DHh + d];
  }
  ssrc[t] = s1; sdst[t] = s2;
}

// per (edge, head): logit + leaky relu + segment max (ordered-int atomicMax)
__global__ void k_edge_logits(const float* __restrict__ he, const float* __restrict__ aedge,
                              const int* __restrict__ ei, const float* __restrict__ ssrc,
                              const float* __restrict__ sdst, float* __restrict__ logit,
                              unsigned* __restrict__ m) {
  int t = blockIdx.x * blockDim.x + threadIdx.x;
  if (t >= Ee * Hh) return;
  int e = t >> 3, hd = t & 7;
  const float* hp = he + (size_t)e * Dd + hd * DHh;
  float se = 0.f;
#pragma unroll
  for (int d = 0; d < DHh; ++d) se += hp[d] * aedge[hd * DHh + d];
  int src = ei[e], dst = ei[Ee + e];
  float l = ssrc[src * Hh + hd] + sdst[dst * Hh + hd] + se;
  l = (l >= 0.f) ? l : 0.2f * l;
  logit[t] = l;
  atomicMax(&m[dst * Hh + hd], f2ord(l));
}

// ex = exp(l - max); segment sum
__global__ void k_exp_sum(float* __restrict__ logit, const unsigned* __restrict__ m,
                          const int* __restrict__ ei, float* __restrict__ ssum) {
  int t = blockIdx.x * blockDim.x + threadIdx.x;
  if (t >= Ee * Hh) return;
  int e = t >> 3, hd = t & 7;
  int dst = ei[Ee + e];
  float mx = ord2f(m[dst * Hh + hd]);
  float ex = __expf(logit[t] - mx);
  logit[t] = ex;  // in-place: logit buffer becomes ex buffer
  atomicAdd(&ssum[dst * Hh + hd], ex);
}

// agg[dst, dim] += alpha * (h[src,dim] + he[e,dim]) ; one thread per (edge, dim)
__global__ void k_aggregate(const float* __restrict__ ex, const float* __restrict__ ssum,
                            const float* __restrict__ h, const float* __restrict__ he,
                            const int* __restrict__ ei, float* __restrict__ agg) {
  int t = blockIdx.x * blockDim.x + threadIdx.x;   // Ee*Dd = 64M < 2^31
  if (t >= Ee * Dd) return;
  int e = t >> 7, dim = t & 127, hd = dim >> 4;
  int src = ei[e], dst = ei[Ee + e];
  float alpha = ex[e * Hh + hd] / (ssum[dst * Hh + hd] + 1e-16f);
  float val = alpha * (h[src * Dd + dim] + he[(size_t)e * Dd + dim]);
  atomicAdd(&agg[dst * Dd + dim], val);
}

__global__ void k_bias_add(float* __restrict__ xo, const float* __restrict__ bias) {
  int t = blockIdx.x * blockDim.x + threadIdx.x;
  if (t >= Nn * Dd) return;
  xo[t] += bias[t & 127];
}

__global__ void k_alpha_mean(const float* __restrict__ ex, const float* __restrict__ ssum,
                             const int* __restrict__ ei, float* __restrict__ am) {
  int e = blockIdx.x * blockDim.x + threadIdx.x;
  if (e >= Ee) return;
  int dst = ei[Ee + e];
  float s = 0.f;
#pragma unroll
  for (int hd = 0; hd < Hh; ++hd) s += ex[e * Hh + hd] / (ssum[dst * Hh + hd] + 1e-16f);
  am[e] = s * 0.125f;
}

// ---------------- top-k via 4096-bin histogram select ----------------------
__global__ void k_hist(const float* __restrict__ am, unsigned* __restrict__ hist) {
  int e = blockIdx.x * blockDim.x + threadIdx.x;
  if (e >= Ee) return;
  atomicAdd(&hist[f2ord(am[e]) >> 20], 1u);
}

__global__ void k_thresh(const unsigned* __restrict__ hist, unsigned* __restrict__ tinfo) {
  unsigned cum = 0;
  int b;
  for (b = 4095; b >= 0; --b) {
    if (cum + hist[b] >= (unsigned)Kk) break;
    cum += hist[b];
  }
  tinfo[0] = (unsigned)b;   // threshold bin
  tinfo[1] = cum;           // count strictly above threshold bin
}

__global__ void k_compact(const float* __restrict__ am, const int* __restrict__ ei,
                          const unsigned* __restrict__ tinfo, unsigned* __restrict__ cnt,
                          int* __restrict__ outi) {
  int e = blockIdx.x * blockDim.x + threadIdx.x;
  if (e >= Ee) return;
  unsigned key = f2ord(am[e]) >> 20;
  unsigned t = tinfo[0], nab = tinfo[1];
  int pos = -1;
  if (key > t) {
    pos = (int)atomicAdd(&cnt[0], 1u);
  } else if (key == t) {
    unsigned p = atomicAdd(&cnt[1], 1u);
    if (p < (unsigned)Kk - nab) pos = (int)(nab + p);
  }
  if (pos >= 0 && pos < Kk) {
    outi[pos]      = ei[e];        // src row
    outi[Kk + pos] = ei[Ee + e];   // dst row
  }
}

// out[0] = f + xa ; out[1] = f + xb
__global__ void k_final(const float* __restrict__ f, const float* __restrict__ xa,
                        const float* __restrict__ xb, float* __restrict__ out) {
  int t = blockIdx.x * blockDim.x + threadIdx.x;
  if (t >= Nn * Dd) return;
  float fv = f[t];
  out[t]           = fv + xa[t];
  out[Nn * Dd + t] = fv + xb[t];
}

// ---------------- host launcher -------------------------------------------
extern "C" void kernel_launch(void* const* d_in, const int* in_sizes, int n_in,
                              void* d_out, int out_size, void* d_ws, size_t ws_size,
                              hipStream_t stream) {
  (void)in_sizes; (void)n_in; (void)out_size;

  // workspace carve-out (branches run sequentially -> reuse big buffers)
  char* ws = (char*)d_ws;
  size_t off = 0;
  auto alloc = [&](size_t bytes) -> char* {
    char* p = ws + off;
    off += (bytes + 255) & ~(size_t)255;
    return p;
  };
  float*    he    = (float*)alloc((size_t)Ee * Dd * 4);   // reused per branch
  float*    hbuf  = (float*)alloc((size_t)Nn * Dd * 4);   // reused per branch
  float*    exbuf = (float*)alloc((size_t)Ee * Hh * 4);   // logits -> ex, reused
  unsigned* mbuf  = (unsigned*)alloc((size_t)Nn * Hh * 4);
  float*    ssum  = (float*)alloc((size_t)Nn * Hh * 4);
  float*    ssrc  = (float*)alloc((size_t)Nn * Hh * 4);
  float*    sdst  = (float*)alloc((size_t)Nn * Hh * 4);
  float*    am    = (float*)alloc((size_t)Ee * 4);
  float*    xa    = (float*)alloc((size_t)Nn * Dd * 4);   // persists
  float*    xb    = (float*)alloc((size_t)Nn * Dd * 4);   // persists
  float*    fbuf  = (float*)alloc((size_t)Nn * Dd * 4);
  float*    scsh  = (float*)alloc(2 * Dd * 4);
  float*    csum  = (float*)alloc(2 * Dd * 4);
  unsigned* hist  = (unsigned*)alloc(4096 * 4);
  unsigned* tinfo = (unsigned*)alloc(2 * 4);
  unsigned* cnt   = (unsigned*)alloc(2 * 4);
  if (off > ws_size) return;  // workspace too small; nothing sane to do

  float* xbr[2] = { xa, xb };
  const int xi[2]  = {0, 1},  eai[2] = {2, 3},   eii[2] = {4, 5};
  const int gi[2]  = {6, 8},  bi[2]  = {7, 9};
  const int Wi[2]  = {10, 16}, Wei[2] = {11, 17};
  const int asi[2] = {12, 18}, adi[2] = {13, 19}, aed[2] = {14, 20}, bsi[2] = {15, 21};

  int* outIdx = (int*)d_out + 2 * Nn * Dd;   // topk indices region (int32 bits)

  for (int br = 0; br < 2; ++br) {
    const float* x     = (const float*)d_in[xi[br]];
    const float* ea    = (const float*)d_in[eai[br]];
    const int*   ei    = (const int*)d_in[eii[br]];
    const float* g     = (const float*)d_in[gi[br]];
    const float* beta  = (const float*)d_in[bi[br]];
    const float* W     = (const float*)d_in[Wi[br]];
    const float* We    = (const float*)d_in[Wei[br]];
    const float* asrc  = (const float*)d_in[asi[br]];
    const float* adst  = (const float*)d_in[adi[br]];
    const float* aedge = (const float*)d_in[aed[br]];
    const float* bias  = (const float*)d_in[bsi[br]];
    float* xo = xbr[br];

    // BatchNorm stats -> affine fold
    hipMemsetAsync(csum, 0, 2 * Dd * 4, stream);
    k_bn_stats<<<512, 128, 0, stream>>>(x, csum);
    k_bn_fin<<<1, 128, 0, stream>>>(csum, g, beta, scsh);

    // WMMA projections
    k_node_proj<<<Nn / 16, 256, 0, stream>>>(x, W, scsh, hbuf);
    k_edge_proj<<<Ee / 16, 256, 0, stream>>>(ea, We, he);

    // attention logits + segment softmax
    k_node_scores<<<(Nn * Hh + 255) / 256, 256, 0, stream>>>(hbuf, asrc, adst, ssrc, sdst);
    hipMemsetAsync(mbuf, 0, (size_t)Nn * Hh * 4, stream);
    hipMemsetAsync(ssum, 0, (size_t)Nn * Hh * 4, stream);
    k_edge_logits<<<(Ee * Hh + 255) / 256, 256, 0, stream>>>(he, aedge, ei, ssrc, sdst, exbuf, mbuf);
    k_exp_sum<<<(Ee * Hh + 255) / 256, 256, 0, stream>>>(exbuf, mbuf, ei, ssum);

    // segment-sum aggregation + bias
    hipMemsetAsync(xo, 0, (size_t)Nn * Dd * 4, stream);
    k_aggregate<<<(Ee * Dd + 255) / 256, 256, 0, stream>>>(exbuf, ssum, hbuf, he, ei, xo);
    k_bias_add<<<(Nn * Dd + 255) / 256, 256, 0, stream>>>(xo, bias);

    // top-k edge select (exact K via histogram threshold)
    k_alpha_mean<<<(Ee + 255) / 256, 256, 0, stream>>>(exbuf, ssum, ei, am);
    hipMemsetAsync(hist, 0, 4096 * 4, stream);
    hipMemsetAsync(cnt, 0, 2 * 4, stream);
    k_hist<<<(Ee + 255) / 256, 256, 0, stream>>>(am, hist);
    k_thresh<<<1, 1, 0, stream>>>(hist, tinfo);
    k_compact<<<(Ee + 255) / 256, 256, 0, stream>>>(am, ei, tinfo, cnt, outIdx + (size_t)br * 2 * Kk);
  }

  // fusion + residuals
  k_fusion<<<Nn / 16, 256, 0, stream>>>(xa, xb, (const float*)d_in[22], (const float*)d_in[23], fbuf);
  k_final<<<(Nn * Dd + 255) / 256, 256, 0, stream>>>(fbuf, xa, xb, (float*)d_out);
}